// BaseHysteresis_66005057405141
// MI455X (gfx1250) — compile-verified
//
#include <hip/hip_runtime.h>
#include <hip/hip_bf16.h>

typedef float v2f __attribute__((ext_vector_type(2)));
typedef float v8f __attribute__((ext_vector_type(8)));

#define BLOCK   256
#define TILE_T  16
#define TILE_S  17          // LDS row stride (floats), coprime 64 -> conflict-free
#define MAX_T   4096
#define TEMP_INV 100.0f     // 1/TEMP, TEMP = 0.01

__device__ __forceinline__ float sigmoidf_fast(float x) {
    return 1.0f / (1.0f + __expf(-x));
}

// Kernel 1: zero per-t accumulator acc[0..T-1]; block 0 also computes
// sum(sigmoid(raw_density)) into acc[T].
__global__ void hyst_init_kernel(const float* __restrict__ raw_density,
                                 float* __restrict__ acc, int T, int n_mesh) {
    int t = blockIdx.x * blockDim.x + threadIdx.x;
    if (t < T) acc[t] = 0.0f;
    if (blockIdx.x == 0) {
        __shared__ float red[BLOCK];
        float s = 0.0f;
        for (int j = threadIdx.x; j < n_mesh; j += BLOCK)
            s += sigmoidf_fast(raw_density[j]);
        red[threadIdx.x] = s;
        __syncthreads();
        for (int off = BLOCK / 2; off > 0; off >>= 1) {
            if (threadIdx.x < off) red[threadIdx.x] += red[threadIdx.x + off];
            __syncthreads();
        }
        if (threadIdx.x == 0) acc[T] = red[0];
    }
}

// Kernel 2: hysteresis scan. One mesh point per thread; h resident in LDS.
// Every 16 timesteps each wave reduces ITS OWN 32 points per-t with 8 chained
// V_WMMA_F32_16X16X4_F32 (B = ones) and atomics into acc[t]. Because a wave
// only reads LDS rows written by its own lanes, and DS ops from one wave are
// processed in order, the main loop needs NO barriers: waves run decoupled.
__global__ void __launch_bounds__(BLOCK)
hyst_scan_kernel(const float* __restrict__ h,
                 const float* __restrict__ mesh_points,
                 const float* __restrict__ raw_density,
                 float* __restrict__ acc,
                 int T, int n_mesh) {
    __shared__ float h_lds[MAX_T];
    __shared__ float tile[BLOCK * TILE_S];   // tile[point * TILE_S + t_local]

    const int tid = threadIdx.x;
    const int j   = blockIdx.x * BLOCK + tid;

    __builtin_prefetch(h, 0, 1);             // global_prefetch_b8 warm-up

    for (int t = tid; t < T && t < MAX_T; t += BLOCK) h_lds[t] = h[t];

    float beta = 0.0f, alpha = 1.0f, d = 0.0f;
    if (j < n_mesh) {
        beta  = mesh_points[2 * j + 0];
        alpha = mesh_points[2 * j + 1];
        d     = sigmoidf_fast(raw_density[j]);
    }
    float s = -1.0f;   // init state
    float f = 0.0f;    // previous h
    __syncthreads();   // only barrier: h_lds visible to all waves

    // WMMA lane mapping for 16x4 f32 A-matrix (per wave):
    // lanes 0-15 hold A[m,0..1], lanes 16-31 hold A[m,2..3]
    const int lane  = tid & 31;
    const int wbase = (tid >> 5) << 5;       // this wave's first mesh point row
    const int half  = (lane >> 4) & 1;
    const int mrow  = lane & 15;

    const int ntiles = T / TILE_T;
    for (int tt = 0; tt < ntiles; ++tt) {
        #pragma unroll
        for (int k = 0; k < TILE_T; ++k) {
            const float ht = h_lds[tt * TILE_T + k];
            const float su = s + ( 1.0f - s) * sigmoidf_fast((ht - alpha) * TEMP_INV);
            const float sd = s + (-1.0f - s) * sigmoidf_fast((beta - ht) * TEMP_INV);
            s = (ht > f) ? su : sd;
            f = ht;
            tile[tid * TILE_S + k] = s * d;   // weighted contribution
        }

        // Per-wave reduction of its own 32 points: EXEC all-ones, no barrier.
        v8f c = {};                      // C/D accumulator (f32, 8 VGPRs)
        v2f b; b.x = 1.0f; b.y = 1.0f;   // B = ones(4x16)
        #pragma unroll
        for (int kb = 0; kb < 8; ++kb) {
            const int p0 = wbase + kb * 4 + half * 2;
            v2f a;
            a.x = tile[(p0 + 0) * TILE_S + mrow];
            a.y = tile[(p0 + 1) * TILE_S + mrow];
            // D = A(16x4) x B(4x16) + C ; accumulates 4 mesh points / issue
            c = __builtin_amdgcn_wmma_f32_16x16x4_f32(
                    false, a, false, b, (short)0, c, false, false);
        }
        // Column N=0 of C lives in lanes 0 (M=0..7) and 16 (M=8..15)
        if ((lane & 15) == 0) {
            const int tbase = tt * TILE_T + half * 8;
            #pragma unroll
            for (int r = 0; r < 8; ++r)
                atomicAdd(&acc[tbase + r], c[r]);
        }
    }
}

// Kernel 3: epilogue  out[t] = scale * acc[t]/sumd + h[t]*slope + offset
__global__ void hyst_final_kernel(const float* __restrict__ h,
                                  const float* __restrict__ raw_offset,
                                  const float* __restrict__ raw_scale,
                                  const float* __restrict__ raw_slope,
                                  const float* __restrict__ acc,
                                  float* __restrict__ out, int T) {
    const int t = blockIdx.x * blockDim.x + threadIdx.x;
    if (t >= T) return;
    const float offset = -10.0f + 20.0f * sigmoidf_fast(raw_offset[0]);
    const float scale  =  20.0f * sigmoidf_fast(raw_scale[0]);
    const float slope  = -20.0f + 40.0f * sigmoidf_fast(raw_slope[0]);
    const float sumd   = acc[T];
    out[t] = scale * (acc[t] / sumd) + h[t] * slope + offset;
}

extern "C" void kernel_launch(void* const* d_in, const int* in_sizes, int n_in,
                              void* d_out, int out_size, void* d_ws, size_t ws_size,
                              hipStream_t stream) {
    const float* h   = (const float*)d_in[0];   // (T,)
    const float* mp  = (const float*)d_in[1];   // (n,2) flat: [beta, alpha]
    const float* rd  = (const float*)d_in[2];   // (n,)
    const float* ro  = (const float*)d_in[3];   // (1,)
    const float* rsc = (const float*)d_in[4];   // (1,)
    const float* rsl = (const float*)d_in[5];   // (1,)
    float* out = (float*)d_out;
    float* acc = (float*)d_ws;                  // T floats (per-t sums) + 1 (sum density)

    const int T      = in_sizes[0];
    const int n_mesh = in_sizes[2];

    const int tb = (T + BLOCK - 1) / BLOCK;
    const int mb = (n_mesh + BLOCK - 1) / BLOCK;

    hipLaunchKernelGGL(hyst_init_kernel,  dim3(tb), dim3(BLOCK), 0, stream,
                       rd, acc, T, n_mesh);
    hipLaunchKernelGGL(hyst_scan_kernel,  dim3(mb), dim3(BLOCK), 0, stream,
                       h, mp, rd, acc, T, n_mesh);
    hipLaunchKernelGGL(hyst_final_kernel, dim3(tb), dim3(BLOCK), 0, stream,
                       h, ro, rsc, rsl, acc, out, T);
}